// SOC_Solver_11751030522578
// MI455X (gfx1250) — compile-verified
//
#include <hip/hip_runtime.h>
#include <math.h>

#define Dd     128
#define Hh     256
#define Bb     512
#define NSTEPS 300
#define NPART  ((NSTEPS + 1) * 16)
#define DT_F   (1.0f / 300.0f)
#define SQDT_F 0.057735026919f   /* sqrt(1/300) */

typedef __attribute__((ext_vector_type(2))) float v2f;
typedef __attribute__((ext_vector_type(8))) float v8f;

__device__ __forceinline__ v8f wmma_f32(v2f a, v2f b, v8f c) {
  // V_WMMA_F32_16X16X4_F32: D = A(16x4) x B(4x16) + C(16x16), all fp32.
  return __builtin_amdgcn_wmma_f32_16x16x4_f32(
      /*neg_a=*/false, a, /*neg_b=*/false, b,
      /*c_mod=*/(short)0, c, /*reuse_a=*/false, /*reuse_b=*/false);
}

// One wave computes a 16x(16*NT) strip of X(rows x K) @ W(K x cols).
// A frag: lane lh (half h) holds X[r0+lh][k+2h], X[r0+lh][k+2h+1].
// B frag (row-major W): rows k+2h, k+2h+1 at column c0+lh.
// B frag (BTR: W stores M row-major, logical W[k][n] = M[n][k]).
template <int NT, int K, bool BTR>
__device__ __forceinline__ void gemm_strip(const float* X, int ldx,
                                           const float* W, int ldw,
                                           int r0, int c0,
                                           int lh, int half, v8f* acc) {
  const float* arow = X + (size_t)(r0 + lh) * ldx + 2 * half;
  for (int k = 0; k < K; k += 4) {
    v2f a;
    a.x = arow[k];
    a.y = arow[k + 1];
#pragma unroll
    for (int t = 0; t < NT; ++t) {
      int col = c0 + 16 * t + lh;
      v2f b;
      if (BTR) {
        const float* bp = W + (size_t)col * ldw + k + 2 * half;
        b.x = bp[0];
        b.y = bp[1];
      } else {
        const float* bp = W + (size_t)(k + 2 * half) * ldw + col;
        b.x = bp[0];
        b.y = bp[ldw];
      }
      acc[t] = wmma_f32(a, b, acc[t]);
    }
  }
}

// ---------------------------------------------------------------------------
// init: states[0] = x0 broadcast; zero log-weight accumulators and output.
// ---------------------------------------------------------------------------
__global__ __launch_bounds__(256) void soc_init(const float* __restrict__ x0,
                                                float* __restrict__ states,
                                                float* __restrict__ lwd,
                                                float* __restrict__ lws,
                                                float* __restrict__ out) {
  int i = blockIdx.x * 256 + threadIdx.x;
  if (i < Bb * Dd) states[i] = x0[i & (Dd - 1)];
  if (i < Bb) { lwd[i] = 0.0f; lws[i] = 0.0f; }
  if (i == 0) out[0] = 0.0f;
}

// ---------------------------------------------------------------------------
// One SDE step, fused: h=tanh([t,x]W1+b1); u=-(hW2+b2); Ax,Px,An,Au GEMMs;
// x_new, g_s, and per-row log-weight updates. 16 WGs x 256 thr, 32 rows each.
// ---------------------------------------------------------------------------
__global__ __launch_bounds__(256) void soc_step(
    const float* __restrict__ W1, const float* __restrict__ b1,
    const float* __restrict__ W2, const float* __restrict__ b2,
    const float* __restrict__ Amat, const float* __restrict__ Pmat,
    const float* __restrict__ noises,
    float* __restrict__ states, float* __restrict__ gbuf,
    float* __restrict__ lw_det, float* __restrict__ lw_sto, int s) {
  __shared__ float u_lds[32][Dd];        // 16 KB
  __shared__ float red[3][32][8];        //  3 KB
  __shared__ float region[32 * Hh];      // 32 KB: h (phases 1-2) / Px+gacc (3-4)
  float* h_lds = region;                 // [32][256]
  float* Px_lds = region;                // [32][128]
  float* gacc = region + 32 * Dd;        // [32][128]

  const int tid = threadIdx.x;
  const int wave = tid >> 5, lane = tid & 31, half = lane >> 4, lh = lane & 15;
  const int rbase = blockIdx.x * 32;
  const float t0 = (float)s * DT_F;

  const float* x = states + (size_t)s * Bb * Dd + (size_t)rbase * Dd;
  const float* nz = noises + (size_t)s * Bb * Dd + (size_t)rbase * Dd;
  float* xnew = states + (size_t)(s + 1) * Bb * Dd + (size_t)rbase * Dd;

  // Phase 1: h = tanh(x @ W1[1:,:] + (b1 + t0*W1[0,:]))
  {
    int m = wave & 1, n0 = (wave >> 1) * 64, r0 = m * 16;
    v8f acc[4] = {};
    gemm_strip<4, Dd, false>(x, Dd, W1 + Hh, Hh, r0, n0, lh, half, acc);
#pragma unroll
    for (int t = 0; t < 4; ++t)
#pragma unroll
      for (int v = 0; v < 8; ++v) {
        int col = n0 + 16 * t + lh;
        int row = r0 + v + 8 * half;
        float c1 = b1[col] + t0 * W1[col];
        h_lds[row * Hh + col] = tanhf(acc[t][v] + c1);
      }
  }
  __syncthreads();
  // Phase 2: u = -(h @ W2 + b2)    (sigma = I)
  {
    int m = wave & 1, n0 = (wave >> 1) * 32, r0 = m * 16;
    v8f acc[2] = {};
    gemm_strip<2, Hh, false>(h_lds, Hh, W2, Dd, r0, n0, lh, half, acc);
#pragma unroll
    for (int t = 0; t < 2; ++t)
#pragma unroll
      for (int v = 0; v < 8; ++v) {
        int col = n0 + 16 * t + lh;
        int row = r0 + v + 8 * half;
        u_lds[row][col] = -(acc[t][v] + b2[col]);
      }
  }
  __syncthreads();
  for (int i = tid; i < 32 * Dd; i += 256) gacc[i] = 0.0f;
  __syncthreads();
  // Phase 3: four 32x128 GEMMs (K=128): Ax, Px, An, Au (B operand transposed)
  {
    int g = wave >> 1, m = wave & 1, r0 = m * 16;
    v8f acc[8] = {};
    const float* Aop = (g <= 1) ? x : (g == 2 ? nz : &u_lds[0][0]);
    const float* Bop = (g == 1) ? Pmat : Amat;
    gemm_strip<8, Dd, true>(Aop, Dd, Bop, Dd, r0, 0, lh, half, acc);
#pragma unroll
    for (int t = 0; t < 8; ++t)
#pragma unroll
      for (int v = 0; v < 8; ++v) {
        int col = 16 * t + lh;
        int row = r0 + v + 8 * half;
        float val = acc[t][v];
        if (g == 0) {  // x_new = x + dt*(Ax + u) + sqrt(dt)*noise
          float xv = x[row * Dd + col];
          float nv = nz[row * Dd + col];
          float uv = u_lds[row][col];
          xnew[row * Dd + col] = xv + DT_F * (val + uv) + SQDT_F * nv;
        } else if (g == 1) {
          Px_lds[row * Dd + col] = val;
        } else if (g == 2) {
          atomicAdd(&gacc[row * Dd + col], -SQDT_F * val);
        } else {
          atomicAdd(&gacc[row * Dd + col], -DT_F * val);
        }
      }
  }
  __syncthreads();
  // Phase 4: g_s = dt*Px + gacc; row reductions for lw_det / lw_sto
  {
    int row = tid >> 3, c0 = (tid & 7) * 16;
    float f = 0.0f, su2 = 0.0f, sun = 0.0f;
#pragma unroll
    for (int j = 0; j < 16; ++j) {
      int col = c0 + j;
      float px = Px_lds[row * Dd + col];
      float uv = u_lds[row][col];
      float nv = nz[row * Dd + col];
      float xv = x[row * Dd + col];
      gbuf[(size_t)s * Bb * Dd + (size_t)(rbase + row) * Dd + col] =
          DT_F * px + gacc[row * Dd + col];
      f += xv * px;
      su2 += uv * uv;
      sun += uv * nv;
    }
    red[0][row][tid & 7] = f;
    red[1][row][tid & 7] = su2;
    red[2][row][tid & 7] = sun;
  }
  __syncthreads();
  if (tid < 32) {
    float f = 0.0f, su2 = 0.0f, sun = 0.0f;
#pragma unroll
    for (int j = 0; j < 8; ++j) {
      f += red[0][tid][j];
      su2 += red[1][tid][j];
      sun += red[2][tid][j];
    }
    lw_det[rbase + tid] += DT_F * (-0.5f * su2 - 0.5f * f);
    lw_sto[rbase + tid] += SQDT_F * sun;
  }
}

// ---------------------------------------------------------------------------
// In-place suffix sum over t of g[t][b][d]; one thread per (b,d).
// ---------------------------------------------------------------------------
__global__ __launch_bounds__(256) void soc_suffix(float* __restrict__ g) {
  int idx = blockIdx.x * 256 + threadIdx.x;  // b*Dd + d
  float run = 0.0f;
  for (int t = NSTEPS - 1; t >= 0; --t) {
    size_t off = (size_t)t * Bb * Dd + idx;
    run += g[off];
    g[off] = run;
  }
}

// ---------------------------------------------------------------------------
// qxT[b] = Q @ x_T[b];  weight[b] = exp(lw_det + lw_sto - 0.5 x_T'Q x_T)
// ---------------------------------------------------------------------------
__global__ __launch_bounds__(256) void soc_weight(
    const float* __restrict__ Qm, const float* __restrict__ states,
    const float* __restrict__ lwd, const float* __restrict__ lws,
    float* __restrict__ qxT, float* __restrict__ wgt) {
  int b = blockIdx.x * 256 + threadIdx.x;
  if (b >= Bb) return;
  const float* xT = states + (size_t)NSTEPS * Bb * Dd + (size_t)b * Dd;
  float xqx = 0.0f;
  for (int i = 0; i < Dd; ++i) {
    float s = 0.0f;
    for (int j = 0; j < Dd; ++j) s += Qm[i * Dd + j] * xT[j];
    qxT[(size_t)b * Dd + i] = s;
    xqx += xT[i] * s;
  }
  wgt[b] = expf(lwd[b] + lws[b] - 0.5f * xqx);
}

// ---------------------------------------------------------------------------
// Objective: nabla_V = MLP([t,x]); lst = cs[t] + Q x_T; sum w*(lst-nabla)^2.
// One WG per (t, 32-row chunk): 301*16 WGs, WMMA MLP, partial per WG.
// ---------------------------------------------------------------------------
__global__ __launch_bounds__(256) void soc_obj(
    const float* __restrict__ W1, const float* __restrict__ b1,
    const float* __restrict__ W2, const float* __restrict__ b2,
    const float* __restrict__ states, const float* __restrict__ cs,
    const float* __restrict__ qxT, const float* __restrict__ wgt,
    float* __restrict__ partials) {
  __shared__ float h_lds[32 * Hh];  // 32 KB
  __shared__ float red[256];

  const int tid = threadIdx.x;
  const int wave = tid >> 5, lane = tid & 31, half = lane >> 4, lh = lane & 15;
  const int bid = blockIdx.x;
  const int t = bid >> 4, chunk = bid & 15;
  const int rbase = chunk * 32;
  const float tt = (float)t * DT_F;
  const float* x = states + (size_t)t * Bb * Dd + (size_t)rbase * Dd;

  {
    int m = wave & 1, n0 = (wave >> 1) * 64, r0 = m * 16;
    v8f acc[4] = {};
    gemm_strip<4, Dd, false>(x, Dd, W1 + Hh, Hh, r0, n0, lh, half, acc);
#pragma unroll
    for (int q = 0; q < 4; ++q)
#pragma unroll
      for (int v = 0; v < 8; ++v) {
        int col = n0 + 16 * q + lh;
        int row = r0 + v + 8 * half;
        float c1 = b1[col] + tt * W1[col];
        h_lds[row * Hh + col] = tanhf(acc[q][v] + c1);
      }
  }
  __syncthreads();
  float part = 0.0f;
  {
    int m = wave & 1, n0 = (wave >> 1) * 32, r0 = m * 16;
    v8f acc[2] = {};
    gemm_strip<2, Hh, false>(h_lds, Hh, W2, Dd, r0, n0, lh, half, acc);
#pragma unroll
    for (int q = 0; q < 2; ++q)
#pragma unroll
      for (int v = 0; v < 8; ++v) {
        int col = n0 + 16 * q + lh;
        int row = r0 + v + 8 * half;
        float nab = acc[q][v] + b2[col];
        float lst = qxT[(size_t)(rbase + row) * Dd + col];
        if (t < NSTEPS)
          lst += cs[(size_t)t * Bb * Dd + (size_t)(rbase + row) * Dd + col];
        float d = lst - nab;
        part += wgt[rbase + row] * d * d;
      }
  }
  red[tid] = part;
  __syncthreads();
  for (int st = 128; st > 0; st >>= 1) {
    if (tid < st) red[tid] += red[tid + st];
    __syncthreads();
  }
  if (tid == 0) partials[bid] = red[0];
}

__global__ __launch_bounds__(256) void soc_reduce(const float* __restrict__ partials,
                                                  int n, float* __restrict__ out) {
  __shared__ float red[256];
  float s = 0.0f;
  for (int i = threadIdx.x; i < n; i += 256) s += partials[i];
  red[threadIdx.x] = s;
  __syncthreads();
  for (int st = 128; st > 0; st >>= 1) {
    if (threadIdx.x < st) red[threadIdx.x] += red[threadIdx.x + st];
    __syncthreads();
  }
  if (threadIdx.x == 0) out[0] = red[0] / (float)((NSTEPS + 1) * Bb);
}

extern "C" void kernel_launch(void* const* d_in, const int* in_sizes, int n_in,
                              void* d_out, int out_size, void* d_ws, size_t ws_size,
                              hipStream_t stream) {
  (void)in_sizes; (void)n_in; (void)out_size; (void)ws_size;
  // setup_inputs order: x0, sigma, A, P, Q, W1, b1, W2, b2, noises
  const float* x0     = (const float*)d_in[0];
  const float* Amat   = (const float*)d_in[2];
  const float* Pmat   = (const float*)d_in[3];
  const float* Qm     = (const float*)d_in[4];
  const float* W1     = (const float*)d_in[5];
  const float* b1     = (const float*)d_in[6];
  const float* W2     = (const float*)d_in[7];
  const float* b2     = (const float*)d_in[8];
  const float* noises = (const float*)d_in[9];

  float* ws = (float*)d_ws;
  size_t off = 0;
  float* states = ws + off;   off += (size_t)(NSTEPS + 1) * Bb * Dd;  // 78.9 MB
  float* gbuf   = ws + off;   off += (size_t)NSTEPS * Bb * Dd;        // 78.6 MB
  float* lwd    = ws + off;   off += Bb;
  float* lws    = ws + off;   off += Bb;
  float* qxT    = ws + off;   off += (size_t)Bb * Dd;
  float* wgt    = ws + off;   off += Bb;
  float* parts  = ws + off;   off += NPART;
  float* out = (float*)d_out;

  soc_init<<<256, 256, 0, stream>>>(x0, states, lwd, lws, out);
  for (int s = 0; s < NSTEPS; ++s)
    soc_step<<<16, 256, 0, stream>>>(W1, b1, W2, b2, Amat, Pmat, noises,
                                     states, gbuf, lwd, lws, s);
  soc_suffix<<<(Bb * Dd) / 256, 256, 0, stream>>>(gbuf);
  soc_weight<<<2, 256, 0, stream>>>(Qm, states, lwd, lws, qxT, wgt);
  soc_obj<<<NPART, 256, 0, stream>>>(W1, b1, W2, b2, states, gbuf, qxT, wgt, parts);
  soc_reduce<<<1, 256, 0, stream>>>(parts, NPART, out);
}